// TotalVariationDenoising_62225486184920
// MI455X (gfx1250) — compile-verified
//
#include <hip/hip_runtime.h>
#include <math.h>

#ifndef __has_builtin
#define __has_builtin(x) 0
#endif

#if __has_builtin(__builtin_amdgcn_tensor_load_to_lds)
#define TVD_HAVE_TDM 1
#else
#define TVD_HAVE_TDM 0
#endif
#if __has_builtin(__builtin_amdgcn_s_wait_tensorcnt)
#define TVD_HAVE_TENSORCNT 1
#else
#define TVD_HAVE_TENSORCNT 0
#endif

#define TVD_B  16
#define TVD_H  512
#define TVD_W  512
#define TVD_HW (TVD_H * TVD_W)
#define TVD_N  (TVD_B * TVD_HW)          // 4194304

#define TILE_W  64
#define TILE_H  16
#define TILE_W1 (TILE_W + 1)
#define TILE_H1 (TILE_H + 1)
#define TILE_ELEMS (TILE_H * TILE_W)
#define DUAL_SMEM_BYTES (TILE_H1 * TILE_W1 * sizeof(float))
#define PRIM_SMEM_BYTES (2 * TILE_ELEMS * sizeof(float))

#define NTILES 4096   // 8 x 32 x 16
#define NBLK 4096
#define NTHR 256

constexpr float TVD_LAM = 1.0f / 4194304.0f;  // lam = 1/n
constexpr float TVD_THR = 0.1f / 4194304.0f;  // reg weight / n (both axes)

typedef unsigned int tvd_u32x4 __attribute__((ext_vector_type(4)));
typedef int          tvd_i32x8 __attribute__((ext_vector_type(8)));
typedef int          tvd_i32x4 __attribute__((ext_vector_type(4)));

#if TVD_HAVE_TDM
// One 2D TDM descriptor: tile_h x tile_w elements (f32) from gsrc (row stride
// `stride` elems) -> LDS byte offset lds_off, rows packed at tile_w stride.
// tensor dims = remaining extent from gsrc so trailing-OOB reads zero-fill.
__device__ __forceinline__ void tvd_tdm_load_2d(const float* gsrc, unsigned lds_off,
                                                unsigned tdim0, unsigned tdim1,
                                                unsigned tile_w, unsigned tile_h,
                                                unsigned stride) {
  unsigned long long ga = (unsigned long long)(uintptr_t)(const void*)gsrc;
  tvd_u32x4 g0 = {
      1u,                                              // count=1, user desc
      lds_off,                                         // lds_addr (bytes)
      (unsigned)(ga & 0xFFFFFFFFu),                    // global_addr[31:0]
      (unsigned)((ga >> 32) & 0x01FFFFFFu) | (2u << 30)  // addr[56:32] | type=2
  };
  tvd_i32x8 g1 = {
      (int)(2u << 16),                                 // data_size = 4B
      (int)((tdim0 & 0xFFFFu) << 16),                  // tensor_dim0[15:0]
      (int)(((tdim0 >> 16) & 0xFFFFu) | ((tdim1 & 0xFFFFu) << 16)),
      (int)(((tdim1 >> 16) & 0xFFFFu) | (tile_w << 16)),  // tile_dim0
      (int)tile_h,                                     // tile_dim1 (tile_dim2=0)
      (int)stride,                                     // tensor_dim0_stride
      0, 0
  };
  tvd_i32x4 g2 = { 1, 1, 0, 0 };                       // tensor_dim2/3 = 1
  tvd_i32x4 g3 = { 0, (int)(1u << 16), 0, 0 };         // tensor_dim4 = 1
  tvd_i32x8 g4 = { 0, 0, 0, 0, 0, 0, 0, 0 };           // VADDR4 group: unused
  __builtin_amdgcn_tensor_load_to_lds(g0, g1, g2, g3, g4, 0);
}
__device__ __forceinline__ void tvd_wait_tensor() {
#if TVD_HAVE_TENSORCNT
  __builtin_amdgcn_s_wait_tensorcnt(0);
#else
  asm volatile("s_wait_tensorcnt 0x0" ::: "memory");
#endif
}
#endif

// ---------------------------------------------------------------------------
// Dual update (first kernel in file -> shown in disasm snippet).
// relaxed tile (17 rows x 65 cols incl. bottom/right halo) DMA'd into LDS by
// the Tensor Data Mover; trailing-OOB reads return zero == reference's
// zero-padded forward difference at the image edges.
// ---------------------------------------------------------------------------
__global__ void k_dual(const float* __restrict__ relaxed, const float* __restrict__ y,
                       float* __restrict__ d0, float* __restrict__ d1,
                       float* __restrict__ d2, const float* __restrict__ sigp) {
  extern __shared__ float tile[];          // 17 x 65 floats, dyn LDS -> offset 0
  const unsigned bx = blockIdx.x;          // flat 1D: 8 x 32 x 16 tiles
  const int w0 = (int)((bx & 7u) << 6);
  const int h0 = (int)(((bx >> 3) & 31u) << 4);
  const int b  = (int)(bx >> 8);
  const int gtile = b * TVD_HW + h0 * TVD_W + w0;

#if TVD_HAVE_TDM
  if (threadIdx.x < 32u) {  // wave 0 only: one DMA per block (TDM ignores EXEC)
    tvd_tdm_load_2d(relaxed + gtile, 0u,
                    (unsigned)(TVD_W - w0), (unsigned)(TVD_H - h0),
                    TILE_W1, TILE_H1, TVD_W);
    tvd_wait_tensor();
  }
#else
  for (int cell = threadIdx.x; cell < TILE_H1 * TILE_W1; cell += NTHR) {
    int r = cell / TILE_W1, c = cell % TILE_W1;
    int gh = h0 + r, gw = w0 + c;
    tile[cell] = (gh < TVD_H && gw < TVD_W) ? relaxed[gtile + r * TVD_W + c] : 0.0f;
  }
#endif
  __syncthreads();

  const float sig = sigp[0];
  const float inv = 1.0f / (sig + TVD_LAM);
#pragma unroll
  for (int e = threadIdx.x; e < TILE_ELEMS; e += NTHR) {
    int r = e >> 6;               // TILE_W == 64
    int c = e & (TILE_W - 1);
    int g = gtile + r * TVD_W + c;
    float ce = tile[r * TILE_W1 + c];
    float ri = tile[r * TILE_W1 + c + 1];        // halo col -> 0 at w==W-1
    float dn = tile[(r + 1) * TILE_W1 + c];      // halo row -> 0 at h==H-1
    float z0 = d0[g] + sig * ce;
    d0[g] = TVD_LAM * (z0 - sig * y[g]) * inv;   // prox_{sigma f0*}
    float z1 = d1[g] + sig * (dn - ce);          // fdiff along H
    d1[g] = fminf(fmaxf(z1, -TVD_THR), TVD_THR);
    float z2 = d2[g] + sig * (ri - ce);          // fdiff along W
    d2[g] = fminf(fmaxf(z2, -TVD_THR), TVD_THR);
  }
}

// ---------------------------------------------------------------------------
// Primal update + over-relaxation (theta=1). dual1/dual2 tiles (16x64) DMA'd
// to LDS (two TDM descriptors, one wait). Leading top/left halo elements are
// read directly from global (64 lanes on the top edge, 16 on the left) since TDM zero-fill only
// applies past the extents, not before the start.
// ---------------------------------------------------------------------------
__global__ void k_primal(const float* __restrict__ d0, const float* __restrict__ d1,
                         const float* __restrict__ d2, float* __restrict__ primal,
                         float* __restrict__ relaxed, const float* __restrict__ sigp) {
  extern __shared__ float lds[];           // 2 x (16 x 64) floats
  float* ldsA = lds;                       // dual1 tile
  float* ldsB = lds + TILE_ELEMS;          // dual2 tile
  const unsigned bx = blockIdx.x;
  const int w0 = (int)((bx & 7u) << 6);
  const int h0 = (int)(((bx >> 3) & 31u) << 4);
  const int b  = (int)(bx >> 8);
  const int gtile = b * TVD_HW + h0 * TVD_W + w0;

#if TVD_HAVE_TDM
  if (threadIdx.x < 32u) {
    tvd_tdm_load_2d(d1 + gtile, 0u,
                    (unsigned)(TVD_W - w0), (unsigned)(TVD_H - h0),
                    TILE_W, TILE_H, TVD_W);
    tvd_tdm_load_2d(d2 + gtile, (unsigned)(TILE_ELEMS * sizeof(float)),
                    (unsigned)(TVD_W - w0), (unsigned)(TVD_H - h0),
                    TILE_W, TILE_H, TVD_W);
    tvd_wait_tensor();                     // in-order per wave: covers both
  }
#else
  for (int e = threadIdx.x; e < TILE_ELEMS; e += NTHR) {
    int off = (e >> 6) * TVD_W + (e & (TILE_W - 1));
    ldsA[e] = d1[gtile + off];
    ldsB[e] = d2[gtile + off];
  }
#endif
  __syncthreads();

  const float tau = sigp[0];
#pragma unroll
  for (int e = threadIdx.x; e < TILE_ELEMS; e += NTHR) {
    int r = e >> 6;
    int c = e & (TILE_W - 1);
    int g = gtile + r * TVD_W + c;
    float d1c = ldsA[e];
    float d2c = ldsB[e];
    float up = (r > 0) ? ldsA[e - TILE_W]
                       : ((h0 > 0) ? d1[g - TVD_W] : 0.0f);  // leading halo
    float lf = (c > 0) ? ldsB[e - 1]
                       : ((w0 > 0) ? d2[g - 1] : 0.0f);
    float kh = d0[g] + (up - d1c) + (lf - d2c);
    float p  = primal[g];
    float pn = p - tau * kh;
    primal[g]  = pn;
    relaxed[g] = 2.0f * pn - p;
  }
}

// ---------------------------------------------------------------------------
// K^H K v  at one point: v + adjH(dH(v)) + adjW(dW(v))   (zero-padded fdiff)
// ---------------------------------------------------------------------------
__device__ __forceinline__ float tvd_khk(const float* __restrict__ v,
                                         int h, int w, int idx) {
  float c  = v[idx];
  float dn = (h < TVD_H - 1) ? v[idx + TVD_W] : 0.0f;
  float up = (h > 0)         ? v[idx - TVD_W] : 0.0f;
  float rt = (w < TVD_W - 1) ? v[idx + 1]     : 0.0f;
  float lf = (w > 0)         ? v[idx - 1]     : 0.0f;
  float dh = dn - c;                                  // fdiff_H at h (dn==0 at edge)
  float ah = ((h > 0) ? (c - up) : 0.0f) - dh;        // adjoint
  float dw = rt - c;
  float aw = ((w > 0) ? (c - lf) : 0.0f) - dw;
  return c + ah + aw;
}

// scalar area init: S[0]=1 (virtual norm for step 0), rest 0
__global__ void k_scal_init(float* S) {
  int t = threadIdx.x;
  if (t < 64) S[t] = (t == 0) ? 1.0f : 0.0f;
}

// deterministic pseudo-random init for power iteration (PCG hash -> [-1,1))
__global__ void k_v0(float* __restrict__ v) {
  int stride = gridDim.x * blockDim.x;
  for (int idx = blockIdx.x * blockDim.x + threadIdx.x; idx < TVD_N; idx += stride) {
    unsigned x = (unsigned)idx * 747796405u + 2891336453u;
    unsigned w = ((x >> ((x >> 28) + 4u)) ^ x) * 277803737u;
    w = (w >> 22) ^ w;
    v[idx] = (float)w * (2.0f / 4294967296.0f) - 1.0f;
  }
}

// one power-iteration step: vout = khk(vin)/sqrt(sumsq_in); partial sums of vout^2
__global__ void k_power(const float* __restrict__ vin, float* __restrict__ vout,
                        const float* __restrict__ sumsq_in,
                        float* __restrict__ partials) {
  __shared__ float sred[NTHR];
  const float s = 1.0f / sqrtf(sumsq_in[0]);   // lazy normalization of vin
  float acc = 0.0f;
  const int stride = gridDim.x * blockDim.x;
  for (int idx = blockIdx.x * blockDim.x + threadIdx.x; idx < TVD_N; idx += stride) {
    int w = idx & (TVD_W - 1);
    int h = (idx >> 9) & (TVD_H - 1);
    float av = tvd_khk(vin, h, w, idx) * s;
    vout[idx] = av;
    acc += av * av;
  }
  int t = threadIdx.x;
  sred[t] = acc;
  __syncthreads();
#pragma unroll
  for (int off = NTHR / 2; off > 0; off >>= 1) {
    if (t < off) sred[t] += sred[t + off];
    __syncthreads();
  }
  if (t == 0) partials[blockIdx.x] = sred[0];
}

// deterministic fixed-order reduction of 4096 partials
__global__ void k_reduce4096(const float* __restrict__ p, float* __restrict__ out) {
  __shared__ float s0[1024];
  int t = threadIdx.x;
  s0[t] = p[t] + p[t + 1024] + p[t + 2048] + p[t + 3072];
  __syncthreads();
#pragma unroll
  for (int off = 512; off > 0; off >>= 1) {
    if (t < off) s0[t] += s0[t + off];
    __syncthreads();
  }
  if (t == 0) out[0] = s0[0];
}

// partial sums of <v, khk v> (dot) and <v,v> (nrm); scale cancels in the ratio
__global__ void k_eig(const float* __restrict__ v,
                      float* __restrict__ pdot, float* __restrict__ pnrm) {
  __shared__ float sd[NTHR];
  __shared__ float sn[NTHR];
  float adot = 0.0f, anrm = 0.0f;
  const int stride = gridDim.x * blockDim.x;
  for (int idx = blockIdx.x * blockDim.x + threadIdx.x; idx < TVD_N; idx += stride) {
    int w = idx & (TVD_W - 1);
    int h = (idx >> 9) & (TVD_H - 1);
    float c = v[idx];
    adot += c * tvd_khk(v, h, w, idx);
    anrm += c * c;
  }
  int t = threadIdx.x;
  sd[t] = adot; sn[t] = anrm;
  __syncthreads();
#pragma unroll
  for (int off = NTHR / 2; off > 0; off >>= 1) {
    if (t < off) { sd[t] += sd[t + off]; sn[t] += sn[t + off]; }
    __syncthreads();
  }
  if (t == 0) { pdot[blockIdx.x] = sd[0]; pnrm[blockIdx.x] = sn[0]; }
}

// sigma = tau = 1/op_norm = sqrt(nrm/dot)
__global__ void k_sigma(const float* __restrict__ pdot, const float* __restrict__ pnrm,
                        float* __restrict__ sig_out) {
  __shared__ float s0[1024];
  __shared__ float s1[1024];
  int t = threadIdx.x;
  s0[t] = pdot[t] + pdot[t + 1024] + pdot[t + 2048] + pdot[t + 3072];
  s1[t] = pnrm[t] + pnrm[t + 1024] + pnrm[t + 2048] + pnrm[t + 3072];
  __syncthreads();
#pragma unroll
  for (int off = 512; off > 0; off >>= 1) {
    if (t < off) { s0[t] += s0[t + off]; s1[t] += s1[t + off]; }
    __syncthreads();
  }
  if (t == 0) sig_out[0] = sqrtf(s1[0] / s0[0]);
}

// primal = relaxed = y ; duals = 0
__global__ void k_state_init(const float* __restrict__ x, float* __restrict__ p,
                             float* __restrict__ r, float* __restrict__ d0,
                             float* __restrict__ d1, float* __restrict__ d2) {
  const int stride = gridDim.x * blockDim.x;
  for (int idx = blockIdx.x * blockDim.x + threadIdx.x; idx < TVD_N; idx += stride) {
    float xv = x[idx];
    p[idx] = xv; r[idx] = xv;
    d0[idx] = 0.0f; d1[idx] = 0.0f; d2[idx] = 0.0f;
  }
}

__global__ void k_copy(const float* __restrict__ src, float* __restrict__ dst, int n) {
  int idx = blockIdx.x * blockDim.x + threadIdx.x;
  if (idx < n) dst[idx] = src[idx];
}

// ---------------------------------------------------------------------------
extern "C" void kernel_launch(void* const* d_in, const int* in_sizes, int n_in,
                              void* d_out, int out_size, void* d_ws, size_t ws_size,
                              hipStream_t stream) {
  const float* x = (const float*)d_in[0];
  float* out = (float*)d_out;

  const size_t need = 65536 + 4ull * TVD_N * sizeof(float);
  if (out_size != TVD_N || in_sizes[0] != TVD_N || ws_size < need) {
    int n = out_size;
    k_copy<<<(n + 255) / 256, 256, 0, stream>>>(x, out, n);
    return;
  }

  float* S     = (float*)d_ws;          // S[0..20] sumsq chain, S[23] sigma
  float* part0 = S + 256;               // 4096 partials
  float* part1 = part0 + 4096;          // 4096 partials
  float* relaxed = (float*)((char*)d_ws + 65536);
  float* dual0   = relaxed + TVD_N;
  float* dual1   = dual0 + TVD_N;
  float* dual2   = dual1 + TVD_N;
  float* vA = dual1;                    // power-iteration buffers reuse dual space
  float* vB = dual2;

  k_scal_init<<<1, 64, 0, stream>>>(S);
  k_v0<<<2048, NTHR, 0, stream>>>(vA);

  for (int i = 0; i < 20; ++i) {
    const float* vin = (i & 1) ? vB : vA;
    float*       vout = (i & 1) ? vA : vB;
    k_power<<<NBLK, NTHR, 0, stream>>>(vin, vout, &S[i], part0);
    k_reduce4096<<<1, 1024, 0, stream>>>(part0, &S[i + 1]);
  }
  // after 20 steps the (unnormalized) eigenvector estimate lives in vA
  k_eig<<<NBLK, NTHR, 0, stream>>>(vA, part0, part1);
  k_sigma<<<1, 1024, 0, stream>>>(part0, part1, &S[23]);

  k_state_init<<<NBLK, NTHR, 0, stream>>>(x, out, relaxed, dual0, dual1, dual2);

  for (int it = 0; it < 20; ++it) {
    k_dual<<<NTILES, NTHR, DUAL_SMEM_BYTES, stream>>>(relaxed, x, dual0, dual1,
                                                      dual2, &S[23]);
    k_primal<<<NTILES, NTHR, PRIM_SMEM_BYTES, stream>>>(dual0, dual1, dual2, out,
                                                        relaxed, &S[23]);
  }
  // d_out == primal after 20 PDHG iterations
}